// SimplePointNet_627065225886
// MI455X (gfx1250) — compile-verified
//
#include <hip/hip_runtime.h>
#include <hip/hip_bf16.h>

#define N_PTS 16384
#define KNN_K 8
#define NBR   (KNN_K + 1)   // 8 nearest + self

typedef __attribute__((ext_vector_type(2))) float v2f;
typedef __attribute__((ext_vector_type(8))) float v8f;

// -------------------------------------------------------------------------
// KNN: one thread per query point, candidates tiled through LDS.
// Register-resident sorted top-8 (ascending); insert is rare-branch guarded.
// -------------------------------------------------------------------------
__global__ __launch_bounds__(256) void knn_kernel(const float* __restrict__ pos,
                                                  int* __restrict__ nbr_idx) {
  __shared__ float sx[256], sy[256], sz[256];
  const int i = blockIdx.x * 256 + threadIdx.x;
  const float px = pos[3 * i + 0];
  const float py = pos[3 * i + 1];
  const float pz = pos[3 * i + 2];
  float bd[KNN_K];
  int   bi[KNN_K];
#pragma unroll
  for (int t = 0; t < KNN_K; ++t) { bd[t] = 3.0e38f; bi[t] = i; }

  for (int base = 0; base < N_PTS; base += 256) {
    __syncthreads();
    {
      const int j = base + threadIdx.x;
      sx[threadIdx.x] = pos[3 * j + 0];
      sy[threadIdx.x] = pos[3 * j + 1];
      sz[threadIdx.x] = pos[3 * j + 2];
    }
    __syncthreads();
#pragma unroll 4
    for (int t = 0; t < 256; ++t) {
      const int j = base + t;
      const float dx = px - sx[t];
      const float dy = py - sy[t];
      const float dz = pz - sz[t];
      const float d = dx * dx + dy * dy + dz * dz;
      if (d < bd[KNN_K - 1] && j != i) {   // rare path
        float dd = d; int jj = j;
#pragma unroll
        for (int p = 0; p < KNN_K; ++p) {  // fully-unrolled sorted insert
          if (dd < bd[p]) {
            const float td = bd[p]; const int ti = bi[p];
            bd[p] = dd; bi[p] = jj;
            dd = td; jj = ti;
          }
        }
      }
    }
  }
#pragma unroll
  for (int t = 0; t < KNN_K; ++t) nbr_idx[i * NBR + t] = bi[t];
  nbr_idx[i * NBR + KNN_K] = i;  // self loop
}

// -------------------------------------------------------------------------
// WMMA helpers: V_WMMA_F32_16X16X4_F32 (fp32 A/B/C, K=4).
// A 16x4 layout: lanes 0-15 -> M=lane, v0=K0,v1=K1 ; lanes 16-31 -> K2,K3.
// B 4x16 layout: lanes 0-15 -> K=k0 (v0) / k0+1 (v1), N=lane ;
//                lanes 16-31 -> K=k0+2 / k0+3, N=lane-16.
// C 16x16: vgpr v: lanes 0-15 -> (M=v, N=lane); lanes 16-31 -> (M=v+8, N=lane-16)
// -------------------------------------------------------------------------
__device__ __forceinline__ v8f wmma_f32_k4(v2f a, v2f b, v8f c) {
  return __builtin_amdgcn_wmma_f32_16x16x4_f32(
      /*neg_a=*/false, a, /*neg_b=*/false, b,
      /*c_mod=*/(short)0, c, /*reuse_a=*/false, /*reuse_b=*/false);
}

__device__ __forceinline__ v2f load_a_frag(const float* __restrict__ m, int ld,
                                           int k0, int lane) {
  const int row = lane & 15;
  const int kb = k0 + ((lane & 16) ? 2 : 0);
  v2f a;
  a.x = m[row * ld + kb];
  a.y = m[row * ld + kb + 1];
  return a;
}

__device__ __forceinline__ v2f load_b_frag(const float* __restrict__ W, int Krows,
                                           int Ncols, int k0, int ntile, int lane) {
  const int col = ntile * 16 + (lane & 15);
  const int kb = k0 + ((lane & 16) ? 2 : 0);
  v2f b;
  b.x = (kb     < Krows) ? W[kb * Ncols + col]       : 0.0f;
  b.y = (kb + 1 < Krows) ? W[(kb + 1) * Ncols + col] : 0.0f;
  return b;
}

// -------------------------------------------------------------------------
// One PointNetConv layer. One wave32 per node; 4 waves (128 thr) per block.
// 16-row edge tile (9 real edges, rows 9..15 duplicate the self edge — max
// aggregation is duplicate-invariant).
//   GEMM1: h = relu(msg @ W1 + b1)            [16 x NH]   (LDS)
//   GEMM2: out = relu(rowmax(h @ W2) + b2)    [NOUT]
// -------------------------------------------------------------------------
template <int KPAD, int KACT, int NH, int NOUT, bool FIRST>
__global__ __launch_bounds__(128) void pointnet_layer_kernel(
    const float* __restrict__ pos, const float* __restrict__ xin,
    const int* __restrict__ nbr_idx,
    const float* __restrict__ W1, const float* __restrict__ b1,
    const float* __restrict__ W2, const float* __restrict__ b2,
    float* __restrict__ xout) {
  __shared__ float s_msg[4][16 * KPAD];
  __shared__ float s_h[4][16 * NH];
  const int lane = threadIdx.x & 31;
  const int wave = threadIdx.x >> 5;
  const int node = blockIdx.x * 4 + wave;
  float* msg = s_msg[wave];
  float* h   = s_h[wave];

  // Build padded 16 x KPAD message matrix (16*KPAD is a multiple of 32 -> no
  // EXEC divergence before the WMMAs).
  for (int e = lane; e < 16 * KPAD; e += 32) {
    const int row = e / KPAD;
    const int col = e - row * KPAD;
    const int r = (row < NBR) ? row : (NBR - 1);
    const int j = nbr_idx[node * NBR + r];
    float val = 0.0f;
    if (FIRST) {
      if (col < 3)      val = pos[3 * j + col];
      else if (col < 6) val = pos[3 * j + (col - 3)] - pos[3 * node + (col - 3)];
    } else {
      if (col < 64)      val = xin[j * 64 + col];
      else if (col < 67) val = pos[3 * j + (col - 64)] - pos[3 * node + (col - 64)];
    }
    msg[row * KPAD + col] = val;
  }
  __syncthreads();

  // GEMM1: 16 x KPAD  @  KACT x NH  -> h (bias + ReLU)
  for (int nt = 0; nt < NH / 16; ++nt) {
    v8f c = {};
#pragma unroll
    for (int k0 = 0; k0 < KPAD; k0 += 4) {
      const v2f a = load_a_frag(msg, KPAD, k0, lane);
      const v2f b = load_b_frag(W1, KACT, NH, k0, nt, lane);
      c = wmma_f32_k4(a, b, c);
    }
    const float bias = b1[nt * 16 + (lane & 15)];
    const int mbase = (lane & 16) ? 8 : 0;
#pragma unroll
    for (int v = 0; v < 8; ++v) {
      h[(mbase + v) * NH + nt * 16 + (lane & 15)] = fmaxf(c[v] + bias, 0.0f);
    }
  }
  __syncthreads();

  // GEMM2: 16 x NH @ NH x NOUT, max over the 16 edge rows, bias, outer ReLU.
  for (int nt = 0; nt < NOUT / 16; ++nt) {
    v8f c = {};
#pragma unroll
    for (int k0 = 0; k0 < NH; k0 += 4) {
      const v2f a = load_a_frag(h, NH, k0, lane);
      const v2f b = load_b_frag(W2, NH, NOUT, k0, nt, lane);
      c = wmma_f32_k4(a, b, c);
    }
    float m = c[0];
#pragma unroll
    for (int v = 1; v < 8; ++v) m = fmaxf(m, c[v]);      // M=0..7 or 8..15
    m = fmaxf(m, __shfl_xor(m, 16, 32));                 // fuse lane halves
    if (lane < 16) {
      xout[node * NOUT + nt * 16 + lane] = fmaxf(m + b2[nt * 16 + lane], 0.0f);
    }
  }
}

// -------------------------------------------------------------------------
// Classifier + log_softmax: one thread per node (128x5 GEMV is negligible).
// -------------------------------------------------------------------------
__global__ __launch_bounds__(256) void classifier_kernel(const float* __restrict__ x,
                                                         const float* __restrict__ Wc,
                                                         const float* __restrict__ bc,
                                                         float* __restrict__ out) {
  const int i = blockIdx.x * 256 + threadIdx.x;
  float acc[5];
#pragma unroll
  for (int c = 0; c < 5; ++c) acc[c] = bc[c];
  for (int k = 0; k < 128; ++k) {
    const float xv = x[i * 128 + k];
#pragma unroll
    for (int c = 0; c < 5; ++c) acc[c] += xv * Wc[k * 5 + c];
  }
  float mx = acc[0];
#pragma unroll
  for (int c = 1; c < 5; ++c) mx = fmaxf(mx, acc[c]);
  float s = 0.0f;
#pragma unroll
  for (int c = 0; c < 5; ++c) s += expf(acc[c] - mx);
  const float ls = logf(s);
#pragma unroll
  for (int c = 0; c < 5; ++c) out[i * 5 + c] = acc[c] - mx - ls;
}

// -------------------------------------------------------------------------
// Launch. Inputs (all fp32): pos, W1a,b1a, W1b,b1b, W2a,b2a, W2b,b2b, Wc,bc
// Workspace: nbr_idx (N*9 i32) | x1 (N*64 f32) | x2 (N*128 f32)  ~= 12.6 MB
// -------------------------------------------------------------------------
extern "C" void kernel_launch(void* const* d_in, const int* in_sizes, int n_in,
                              void* d_out, int out_size, void* d_ws, size_t ws_size,
                              hipStream_t stream) {
  const float* pos = (const float*)d_in[0];
  const float* W1a = (const float*)d_in[1];
  const float* b1a = (const float*)d_in[2];
  const float* W1b = (const float*)d_in[3];
  const float* b1b = (const float*)d_in[4];
  const float* W2a = (const float*)d_in[5];
  const float* b2a = (const float*)d_in[6];
  const float* W2b = (const float*)d_in[7];
  const float* b2b = (const float*)d_in[8];
  const float* Wc  = (const float*)d_in[9];
  const float* bc  = (const float*)d_in[10];
  float* out = (float*)d_out;

  char* ws = (char*)d_ws;
  int*   nbr = (int*)ws;                                        // N*9 ints
  float* x1  = (float*)(ws + (size_t)N_PTS * NBR * sizeof(int));// N*64 f32
  float* x2  = x1 + (size_t)N_PTS * 64;                         // N*128 f32

  knn_kernel<<<N_PTS / 256, 256, 0, stream>>>(pos, nbr);

  // Layer 1: msg K=6 (pad 8) -> 32 -> 64
  pointnet_layer_kernel<8, 6, 32, 64, true><<<N_PTS / 4, 128, 0, stream>>>(
      pos, nullptr, nbr, W1a, b1a, W1b, b1b, x1);

  // Layer 2: msg K=67 (pad 68) -> 64 -> 128
  pointnet_layer_kernel<68, 67, 64, 128, false><<<N_PTS / 4, 128, 0, stream>>>(
      pos, x1, nbr, W2a, b2a, W2b, b2b, x2);

  classifier_kernel<<<N_PTS / 256, 256, 0, stream>>>(x2, Wc, bc, out);
}